// GCNLayer_33062658245473
// MI455X (gfx1250) — compile-verified
//
#include <hip/hip_runtime.h>

// Problem constants (match reference)
#define N_NODES 100000
#define N_EDGES 640000
#define FEATS   128

typedef __attribute__((ext_vector_type(2))) float v2f;
typedef __attribute__((ext_vector_type(8))) float v8f;

typedef __attribute__((address_space(1))) int gas_i32;  // global
typedef __attribute__((address_space(3))) int las_i32;  // LDS

// W stored in LDS k-pair interleaved: Wp[k/2][n] = {W[k][n], W[k+1][n]}.
// Pair-row stride 288 floats: 288 % 64 == 32 -> the two lane-halves of a wave
// (which read pair-rows rp and rp+1) hit disjoint 32-bank sets.
#define P_STRIDE 288
#define W_LDS_BYTES (64 * P_STRIDE * 4)   // 73728 B

// ---------------------------------------------------------------------------
// Kernel 1: zero the h accumulator workspace (float4 stores)
// ---------------------------------------------------------------------------
__global__ void gcn_zero_h(float4* __restrict__ h4, int n4) {
    int i = blockIdx.x * blockDim.x + threadIdx.x;
    if (i < n4) h4[i] = make_float4(0.f, 0.f, 0.f, 0.f);
}

// ---------------------------------------------------------------------------
// Kernel 2: edge scatter-add. One wave32 per edge; each lane owns 4 feats.
// feature (51.2 MB) and h (51.2 MB) are L2-resident (192 MB), so the gathers
// are cheap and the f32 atomics resolve at L2.
// ---------------------------------------------------------------------------
__global__ __launch_bounds__(256)
void gcn_scatter_add(const float* __restrict__ feat,
                     const int* __restrict__ src,
                     const int* __restrict__ dst,
                     float* __restrict__ h) {
    int e    = (blockIdx.x * blockDim.x + threadIdx.x) >> 5;
    int lane = threadIdx.x & 31;
    if (e >= N_EDGES) return;

    int s = src[e];
    int d = dst[e];

    const float4* frow = (const float4*)(feat + (size_t)s * FEATS);
    float4 v = frow[lane];

    float* hrow = h + (size_t)d * FEATS + lane * 4;
    atomicAdd(hrow + 0, v.x);
    atomicAdd(hrow + 1, v.y);
    atomicAdd(hrow + 2, v.z);
    atomicAdd(hrow + 3, v.w);
}

// ---------------------------------------------------------------------------
// Kernel 3: out = h @ W + b via V_WMMA_F32_16X16X4_F32.
// Block = 256 threads = 8 waves; block covers 128 rows, wave w owns rows
// [blk*128 + w*16, +16) and all 8 N-tiles (acc = 8 x v8f).
//
// Fragment layouts per CDNA5 ISA (16x16x4 f32):
//   A: lane l -> row m = l&15, k = (l>>4)*2 + {0,1}   (v2f, global_load_b64)
//   B: lane l -> col n = l&15, k = (l>>4)*2 + {0,1}   (v2f, one ds_load_b64
//      thanks to the k-pair-interleaved LDS image of W)
//   D: VGPR v -> row v + (l>>4)*8, col l&15           (v8f)
// ---------------------------------------------------------------------------
__global__ __launch_bounds__(256)
void gcn_gemm_bias(const float* __restrict__ h,
                   const float* __restrict__ W,
                   const float* __restrict__ bias,
                   float* __restrict__ out) {
    extern __shared__ float Ws[];  // [64 pair-rows][P_STRIDE]

    const int tid = threadIdx.x;

    // ---- Stage W[128][128] -> LDS, k-pair interleaved -------------------
    // Thread t owns column n = t&127 and k-half t>>7; per k iteration the
    // 128 active source addresses are contiguous (coalesced 512B bursts).
    {
        const int n  = tid & 127;
        const int kh = tid >> 7;                 // 0 or 1
        for (int k = kh * 64; k < kh * 64 + 64; ++k) {
            const int dest = (k >> 1) * P_STRIDE + 2 * n + (k & 1);
#if __has_builtin(__builtin_amdgcn_global_load_async_to_lds_b32)
            // CDNA5 async DMA (ASYNCcnt): global -> LDS without VGPR bounce.
            __builtin_amdgcn_global_load_async_to_lds_b32(
                (gas_i32*)(W + (size_t)k * FEATS + n),
                (las_i32*)(Ws + dest), 0, 0);
#else
            Ws[dest] = W[(size_t)k * FEATS + n];
#endif
        }
    }
#if __has_builtin(__builtin_amdgcn_global_load_async_to_lds_b32)
    asm volatile("s_wait_asynccnt 0x0" ::: "memory");
#endif
    __syncthreads();

    const int wave  = tid >> 5;
    const int lane  = tid & 31;
    const int khalf = lane >> 4;   // 0: k in {0,1}; 1: k in {2,3}
    const int nlane = lane & 15;

    const int rowbase = blockIdx.x * 128 + wave * 16;

    // A row for this lane; clamp so out-of-range blocks still keep EXEC all-1s
    int m  = rowbase + nlane;
    int mc = (m < N_NODES) ? m : (N_NODES - 1);
    const float* arow = h + (size_t)mc * FEATS;
    __builtin_prefetch(arow, 0, 3);

    v8f acc[8] = {};

    for (int kk = 0; kk < FEATS; kk += 4) {
        const int k0 = kk + khalf * 2;           // even
        // 8-byte aligned -> global_load_b64
        v2f a = *(const v2f*)(arow + k0);

        // pair-row for this lane-half: rp = k0/2 = kk/2 + khalf
        const float* wrow = Ws + (size_t)(k0 >> 1) * P_STRIDE;
#pragma unroll
        for (int nt = 0; nt < 8; ++nt) {
            const int n = nt * 16 + nlane;
            // {W[k0][n], W[k0+1][n]} contiguous -> single ds_load_b64
            v2f b = *(const v2f*)(wrow + 2 * n);
            acc[nt] = __builtin_amdgcn_wmma_f32_16x16x4_f32(
                /*neg_a=*/false, a, /*neg_b=*/false, b,
                /*c_mod=*/(short)0, acc[nt],
                /*reuse_a=*/false, /*reuse_b=*/false);
        }
    }

    // ---- Epilogue -------------------------------------------------------
    float bv[8];
#pragma unroll
    for (int nt = 0; nt < 8; ++nt) bv[nt] = bias[nt * 16 + nlane];

    if (rowbase + 16 <= N_NODES) {
        // Fast path (all blocks but the last): wave-uniform branch,
        // unpredicated non-temporal stores (out is write-once).
        float* orow = out + (size_t)(rowbase + khalf * 8) * FEATS + nlane;
#pragma unroll
        for (int v = 0; v < 8; ++v) {
            float* p = orow + (size_t)v * FEATS;
#pragma unroll
            for (int nt = 0; nt < 8; ++nt) {
                __builtin_nontemporal_store(acc[nt][v] + bv[nt], p + nt * 16);
            }
        }
    } else {
        // Tail block: per-row predication.
#pragma unroll
        for (int v = 0; v < 8; ++v) {
            const int row = rowbase + khalf * 8 + v;
            if (row < N_NODES) {
                float* p = out + (size_t)row * FEATS + nlane;
#pragma unroll
                for (int nt = 0; nt < 8; ++nt) {
                    __builtin_nontemporal_store(acc[nt][v] + bv[nt], p + nt * 16);
                }
            }
        }
    }
}

// ---------------------------------------------------------------------------
// Host launcher
// ---------------------------------------------------------------------------
extern "C" void kernel_launch(void* const* d_in, const int* in_sizes, int n_in,
                              void* d_out, int out_size, void* d_ws, size_t ws_size,
                              hipStream_t stream) {
    const float* feature = (const float*)d_in[0];
    const int*   edge    = (const int*)d_in[1];   // [2, N_EDGES]: src row then dst row
    const float* W       = (const float*)d_in[2];
    const float* bias    = (const float*)d_in[3];
    float*       out     = (float*)d_out;
    float*       h       = (float*)d_ws;          // N_NODES * FEATS floats

    (void)in_sizes; (void)n_in; (void)out_size; (void)ws_size;

    // 1) h = 0
    const int n4 = N_NODES * FEATS / 4;
    gcn_zero_h<<<(n4 + 255) / 256, 256, 0, stream>>>((float4*)h, n4);

    // 2) h[dst] += feature[src]  (one wave per edge)
    const int scatter_threads = N_EDGES * 32;
    gcn_scatter_add<<<(scatter_threads + 255) / 256, 256, 0, stream>>>(
        feature, edge, edge + N_EDGES, h);

    // 3) out = h @ W + bias (WMMA f32)
    const int gemm_blocks = (N_NODES + 127) / 128;  // 782
    gcn_gemm_bias<<<gemm_blocks, 256, W_LDS_BYTES, stream>>>(h, W, bias, out);
}